// MultiScaleDeformableCorrelation_45664092291269
// MI455X (gfx1250) — compile-verified
//
#include <hip/hip_runtime.h>

typedef __attribute__((ext_vector_type(2))) float v2f;
typedef __attribute__((ext_vector_type(8))) float v8f;

__device__ __forceinline__ int iclamp(int v, int lo, int hi) {
  return v < lo ? lo : (v > hi ? hi : v);
}

// Generic bilinear resize (align_corners=False, PyTorch semantics).
// Input plane index:  pair_mode ? (q>>1)*in_mul + (q&1) + in_off : q*in_mul + in_off
// Output plane index: q*out_mul + out_off          (plane size = hout*wout)
__global__ void __launch_bounds__(256) msdc_resize_bilinear(
    const float* __restrict__ in, float* __restrict__ out,
    int P, int hin, int win, int hout, int wout,
    int pair_mode, int in_mul, int in_off,
    int out_mul, int out_off, float gain)
{
  long idx = (long)blockIdx.x * blockDim.x + threadIdx.x;
  long planesz = (long)hout * wout;
  long total = (long)P * planesz;
  if (idx >= total) return;
  int q   = (int)(idx / planesz);
  int rem = (int)(idx % planesz);
  int oy = rem / wout;
  int ox = rem % wout;

  long ip = pair_mode ? ((long)(q >> 1) * in_mul + (q & 1) + in_off)
                      : ((long)q * in_mul + in_off);
  const float* src = in + ip * (long)hin * win;

  float sy = (oy + 0.5f) * ((float)hin / (float)hout) - 0.5f;
  sy = fminf(fmaxf(sy, 0.f), (float)(hin - 1));
  int y0 = (int)floorf(sy);
  int y1 = (y0 + 1 < hin) ? y0 + 1 : hin - 1;
  float wy = sy - (float)y0;

  float sx = (ox + 0.5f) * ((float)win / (float)wout) - 0.5f;
  sx = fminf(fmaxf(sx, 0.f), (float)(win - 1));
  int x0 = (int)floorf(sx);
  int x1 = (x0 + 1 < win) ? x0 + 1 : win - 1;
  float wx = sx - (float)x0;

  float v00 = src[(long)y0 * win + x0];
  float v01 = src[(long)y0 * win + x1];
  float v10 = src[(long)y1 * win + x0];
  float v11 = src[(long)y1 * win + x1];
  float v = (1.f - wy) * ((1.f - wx) * v00 + wx * v01)
          +         wy * ((1.f - wx) * v10 + wx * v11);

  long op = (long)q * out_mul + out_off;
  out[op * planesz + rem] = v * gain;
}

// Deformable correlation: one wave32 per 16-pixel strip.
// Channel dots computed with V_WMMA_F32_16X16X4_F32; diag(D) = per-pixel dot.
// A layout (16x4 f32): lanes 0-15 hold K=0,1 in v[0],v[1]; lanes 16-31 K=2,3.
// C/H/W are compile-time so all plane strides fold into immediate offsets /
// constant-stride pointer increments (no per-iteration 64-bit multiplies).
// Grid is sized so every wave maps to a valid strip (EXEC all ones for WMMA).
template <int C, int H, int W>
__global__ void __launch_bounds__(256) msdc_deform_corr_wmma(
    const float* __restrict__ f1, const float* __restrict__ f2,
    const float* __restrict__ off, float* __restrict__ corr,
    int off_bstride, int out_mul, int out_off)
{
  __shared__ float diag_tile[8][16][16];

  constexpr long HW   = (long)H * W;
  constexpr long STEP = 4 * HW;       // elements per K=4 channel chunk
  constexpr int  spr  = W >> 4;       // strips per row
  constexpr int  spi  = H * spr;      // strips per image (power of two)

  const int lane = threadIdx.x & 31;
  const int wv   = threadIdx.x >> 5;
  long strip = (long)blockIdx.x * 8 + wv;
  int b   = (int)(strip / spi);
  int rem = (int)(strip % spi);
  int y   = rem / spr;
  int x0s = (rem % spr) << 4;

  const int p     = lane & 15;    // pixel within strip (both half-waves)
  const int chsel = lane >> 4;    // which channel pair of the K=4 chunk
  const int x     = x0s + p;

  const float* f1b  = f1  + (long)b * C * HW;
  const float* f2b  = f2  + (long)b * C * HW;
  const float* offb = off + (long)b * off_bstride * HW;

  long apix = (long)y * W + x;
  float dx = offb[apix];
  float dy = offb[HW + apix];
  float sx = (float)x + dx;
  float sy = (float)y + dy;
  float fx = floorf(sx), fy = floorf(sy);
  float wx1 = sx - fx, wy1 = sy - fy;
  float wx0 = 1.f - wx1, wy0 = 1.f - wy1;
  int ix0 = (int)fx, iy0 = (int)fy;
  int ix1 = ix0 + 1, iy1 = iy0 + 1;
  float vx0 = (ix0 >= 0 && ix0 < W) ? 1.f : 0.f;
  float vx1 = (ix1 >= 0 && ix1 < W) ? 1.f : 0.f;
  float vy0 = (iy0 >= 0 && iy0 < H) ? 1.f : 0.f;
  float vy1 = (iy1 >= 0 && iy1 < H) ? 1.f : 0.f;
  float w00 = wy0 * wx0 * vy0 * vx0;
  float w01 = wy0 * wx1 * vy0 * vx1;
  float w10 = wy1 * wx0 * vy1 * vx0;
  float w11 = wy1 * wx1 * vy1 * vx1;
  int cx0 = iclamp(ix0, 0, W - 1), cx1 = iclamp(ix1, 0, W - 1);
  int cy0 = iclamp(iy0, 0, H - 1), cy1 = iclamp(iy1, 0, H - 1);
  long a00 = (long)cy0 * W + cx0;
  long a01 = (long)cy0 * W + cx1;
  long a10 = (long)cy1 * W + cx0;
  long a11 = (long)cy1 * W + cx1;

  // Loop-carried base pointers; advance by constant STEP each chunk.
  const long cb = (long)(2 * chsel) * HW;
  const float* f1p = f1b + cb + apix;
  const float* p00 = f2b + cb + a00;
  const float* p01 = f2b + cb + a01;
  const float* p10 = f2b + cb + a10;
  const float* p11 = f2b + cb + a11;

  v8f acc = {};
#pragma unroll 4
  for (int c = 0; c < C; c += 4) {
    __builtin_prefetch(f1p + STEP, 0, 1);   // global_prefetch_b8 (speculative)
    __builtin_prefetch(p00 + STEP, 0, 1);
    v2f a, bm;
    a.x = f1p[0];
    a.y = f1p[HW];
    bm.x = w00 * p00[0]  + w01 * p01[0]  + w10 * p10[0]  + w11 * p11[0];
    bm.y = w00 * p00[HW] + w01 * p01[HW] + w10 * p10[HW] + w11 * p11[HW];
    acc = __builtin_amdgcn_wmma_f32_16x16x4_f32(
        /*neg_a=*/false, a, /*neg_b=*/false, bm,
        /*c_mod=*/(short)0, acc, /*reuse_a=*/false, /*reuse_b=*/false);
    f1p += STEP; p00 += STEP; p01 += STEP; p10 += STEP; p11 += STEP;
  }

  // Spill the 16x16 D tile to LDS; diagonal holds the per-pixel dots.
#pragma unroll
  for (int r = 0; r < 8; ++r)
    diag_tile[wv][chsel * 8 + r][p] = acc[r];
  __builtin_amdgcn_wave_barrier();
  asm volatile("s_wait_dscnt 0x0" ::: "memory");
  __builtin_amdgcn_wave_barrier();

  if (lane < 16) {
    float v = diag_tile[wv][lane][lane];
    float* cp = corr + (long)(b * out_mul + out_off) * HW;
    cp[(long)y * W + x0s + lane] = v;
  }
}

extern "C" void kernel_launch(void* const* d_in, const int* in_sizes, int n_in,
                              void* d_out, int out_size, void* d_ws, size_t ws_size,
                              hipStream_t stream) {
  (void)in_sizes; (void)n_in; (void)out_size; (void)ws_size;
  const float* feat1 = (const float*)d_in[0];
  const float* feat2 = (const float*)d_in[1];
  const float* offs  = (const float*)d_in[2];
  float* out = (float*)d_out;

  const int B = 4, C = 128, H = 256, W = 256;
  const int H1 = 128, W1 = 128, H2 = 64, W2 = 64;

  float* ws = (float*)d_ws;
  size_t n_f_s1 = (size_t)B * C * H1 * W1;
  size_t n_f_s2 = (size_t)B * C * H2 * W2;
  size_t n_o_s1 = (size_t)B * 2 * H1 * W1;
  size_t n_o_s2 = (size_t)B * 2 * H2 * W2;
  size_t n_c_s1 = (size_t)B * H1 * W1;

  float* f1s1  = ws;              ws += n_f_s1;
  float* f2s1  = ws;              ws += n_f_s1;
  float* f1s2  = ws;              ws += n_f_s2;
  float* f2s2  = ws;              ws += n_f_s2;
  float* offs1 = ws;              ws += n_o_s1;
  float* offs2 = ws;              ws += n_o_s2;
  float* corr1 = ws;              ws += n_c_s1;
  float* corr2 = ws;

  auto nb = [](size_t n) { return (unsigned)((n + 255) / 256); };

  // --- build pyramids (direct resize from full res, matching reference) ---
  msdc_resize_bilinear<<<nb(n_f_s1), 256, 0, stream>>>(feat1, f1s1, B * C, H, W, H1, W1, 0, 1, 0, 1, 0, 1.f);
  msdc_resize_bilinear<<<nb(n_f_s1), 256, 0, stream>>>(feat2, f2s1, B * C, H, W, H1, W1, 0, 1, 0, 1, 0, 1.f);
  msdc_resize_bilinear<<<nb(n_f_s2), 256, 0, stream>>>(feat1, f1s2, B * C, H, W, H2, W2, 0, 1, 0, 1, 0, 1.f);
  msdc_resize_bilinear<<<nb(n_f_s2), 256, 0, stream>>>(feat2, f2s2, B * C, H, W, H2, W2, 0, 1, 0, 1, 0, 1.f);
  // offsets: slice channels [2i,2i+1] out of (B,6,H,W), resize, scale by 1/2^s
  msdc_resize_bilinear<<<nb(n_o_s1), 256, 0, stream>>>(offs, offs1, B * 2, H, W, H1, W1, 1, 6, 2, 1, 0, 0.5f);
  msdc_resize_bilinear<<<nb(n_o_s2), 256, 0, stream>>>(offs, offs2, B * 2, H, W, H2, W2, 1, 6, 4, 1, 0, 0.25f);

  // --- deformable correlation (WMMA) ---
  // strips = B*H*(W/16); 8 waves/block; all counts divide exactly -> full waves
  unsigned blk0 = (unsigned)((size_t)B * H  * (W  / 16) / 8);
  unsigned blk1 = (unsigned)((size_t)B * H1 * (W1 / 16) / 8);
  unsigned blk2 = (unsigned)((size_t)B * H2 * (W2 / 16) / 8);
  // scale 0: offsets come straight from input (per-batch stride 6 planes),
  //          corr written straight into output plane b*3+0
  msdc_deform_corr_wmma<128, 256, 256><<<blk0, 256, 0, stream>>>(
      feat1, feat2, offs, out, 6, 3, 0);
  msdc_deform_corr_wmma<128, 128, 128><<<blk1, 256, 0, stream>>>(
      f1s1, f2s1, offs1, corr1, 2, 1, 0);
  msdc_deform_corr_wmma<128, 64, 64><<<blk2, 256, 0, stream>>>(
      f1s2, f2s2, offs2, corr2, 2, 1, 0);

  // --- upsample coarse correlations into output planes b*3+1, b*3+2 ---
  size_t n_up = (size_t)B * H * W;
  msdc_resize_bilinear<<<nb(n_up), 256, 0, stream>>>(corr1, out, B, H1, W1, H, W, 0, 1, 0, 3, 1, 1.f);
  msdc_resize_bilinear<<<nb(n_up), 256, 0, stream>>>(corr2, out, B, H2, W2, H, W, 0, 1, 0, 3, 2, 1.f);
}